// dynedgeEdgepool_23029614641731
// MI455X (gfx1250) — compile-verified
//
#include <hip/hip_runtime.h>
#include <math.h>

#define G_ 128
#define P_ 256
#define K_ 4
#define N_ (G_*P_)       // 32768
#define NK_ (N_*K_)      // 131072
#define HCAT_ 772        // concat width: 4 + 4*192
#define C2_ 192          // EdgeConv output width

typedef float v2f __attribute__((ext_vector_type(2)));
typedef float v8f __attribute__((ext_vector_type(8)));

__device__ __forceinline__ float lrelu(float x) { return x > 0.f ? x : 0.01f * x; }

// ------------------------------------------------------------------
// KNN: one block per graph, one thread per target node. Positions in
// LDS; keep 4 smallest d2 (strict < keeps earliest index on ties,
// matching lax.top_k). Invalid key nodes excluded.
// ------------------------------------------------------------------
template<int D>
__global__ void knn_kernel(const float* pos, int stride, int off,
                           const int* valid, int* __restrict__ nbr)
{
    __shared__ float sp[P_ * 3];
    __shared__ int   sv[P_];
    const int g = blockIdx.x, t = threadIdx.x;
    const int gp = g * P_ + t;
    #pragma unroll
    for (int d = 0; d < 3; ++d)
        sp[t*3 + d] = (d < D) ? pos[(size_t)gp * stride + off + d] : 0.f;
    sv[t] = valid ? valid[gp] : 1;
    __syncthreads();

    float bd[K_]; int bi[K_];
    #pragma unroll
    for (int k = 0; k < K_; ++k) { bd[k] = __builtin_inff(); bi[k] = 0; }
    const float px = sp[t*3+0], py = sp[t*3+1], pz = sp[t*3+2];
    for (int j = 0; j < P_; ++j) {
        if (j == t || !sv[j]) continue;
        float dx = sp[j*3+0]-px, dy = sp[j*3+1]-py, dz = sp[j*3+2]-pz;
        float d2 = dx*dx + dy*dy + dz*dz;
        if (d2 < bd[K_-1]) {
            int ins = K_-1;
            while (ins > 0 && d2 < bd[ins-1]) { bd[ins]=bd[ins-1]; bi[ins]=bi[ins-1]; --ins; }
            bd[ins] = d2; bi[ins] = j;
        }
    }
    #pragma unroll
    for (int k = 0; k < K_; ++k) nbr[gp*K_ + k] = bi[k];
}

// ------------------------------------------------------------------
// EdgePooling scores: raw = [x_j, x_i] @ ep_w + ep_b; softmax over K
// in-edges per target; +0.5.
// ------------------------------------------------------------------
__global__ void ep_score_kernel(const float* __restrict__ x, const int* __restrict__ nbr0,
                                const float* __restrict__ ep_w, const float* __restrict__ ep_b,
                                float* __restrict__ escore)
{
    int i = blockIdx.x * blockDim.x + threadIdx.x;
    if (i >= N_) return;
    int g = i >> 8;
    float w[8];
    #pragma unroll
    for (int f = 0; f < 8; ++f) w[f] = ep_w[f];
    float b = ep_b[0];
    float xi0 = x[i*4+0], xi1 = x[i*4+1], xi2 = x[i*4+2], xi3 = x[i*4+3];
    float xiw = xi0*w[4] + xi1*w[5] + xi2*w[6] + xi3*w[7];
    float raw[K_];
    #pragma unroll
    for (int k = 0; k < K_; ++k) {
        int j = (g << 8) + nbr0[i*K_ + k];
        raw[k] = b + xiw + x[j*4+0]*w[0] + x[j*4+1]*w[1] + x[j*4+2]*w[2] + x[j*4+3]*w[3];
    }
    float m = raw[0];
    #pragma unroll
    for (int k = 1; k < K_; ++k) m = fmaxf(m, raw[k]);
    float ek[K_], s = 0.f;
    #pragma unroll
    for (int k = 0; k < K_; ++k) { ek[k] = expf(raw[k] - m); s += ek[k]; }
    #pragma unroll
    for (int k = 0; k < K_; ++k) escore[i*K_ + k] = ek[k] / s + 0.5f;
}

// ------------------------------------------------------------------
// EdgePooling greedy match. Edges never cross graphs, so the global
// stable argsort decomposes per graph: bitonic sort 1024 edges in LDS
// (score desc, flat-id asc on ties == stable sort of -score), then a
// serial greedy match by thread 0. Writes merged x into hcat[:, 0:4].
// ------------------------------------------------------------------
__global__ void ep_match_kernel(const float* __restrict__ x, const int* __restrict__ nbr0,
                                const float* __restrict__ escore,
                                float* hcat, int* __restrict__ validb)
{
    __shared__ float ss[P_*K_];
    __shared__ int   sid[P_*K_];
    __shared__ int   snbr[P_*K_];
    __shared__ int   matched[P_];
    __shared__ int   partner[P_];
    __shared__ float cscale[P_];
    const int g = blockIdx.x, t = threadIdx.x;
    const int E = P_ * K_;   // 1024, blockDim.x == 1024

    ss[t]   = escore[g*E + t];
    sid[t]  = t;
    snbr[t] = nbr0[g*E + t];
    if (t < P_) { matched[t] = 0; partner[t] = t; cscale[t] = 1.0f; }
    __syncthreads();

    for (int ksz = 2; ksz <= E; ksz <<= 1) {
        for (int j = ksz >> 1; j > 0; j >>= 1) {
            int i = t, ixj = i ^ j;
            if (ixj > i) {
                float s1 = ss[i], s2 = ss[ixj];
                int i1 = sid[i], i2 = sid[ixj];
                bool before = (s1 > s2) || (s1 == s2 && i1 < i2);
                bool up = ((i & ksz) == 0);
                if (before != up) { ss[i]=s2; ss[ixj]=s1; sid[i]=i2; sid[ixj]=i1; }
            }
            __syncthreads();
        }
    }

    if (t == 0) {
        for (int e = 0; e < E; ++e) {
            int id = sid[e];
            int d  = id >> 2;        // target node (local)
            int s  = snbr[id];       // source node (local)
            if (!matched[s] && !matched[d]) {
                matched[s] = 1; matched[d] = 1;
                partner[s] = d; partner[d] = s;
                int r = s < d ? s : d;
                cscale[r] = ss[e];
            }
        }
    }
    __syncthreads();

    if (t < P_) {
        int p = t, gp = g*P_ + p;
        int m = matched[p], q = partner[p];
        bool isrep = (!m) || (p < q);
        float sc = (m && isrep) ? cscale[p] : 1.0f;
        #pragma unroll
        for (int f = 0; f < 4; ++f) {
            float v = 0.f;
            if (isrep) {
                v = x[gp*4 + f];
                if (m) v += x[(g*P_ + q)*4 + f];
                v *= sc;
            }
            hcat[(size_t)gp * HCAT_ + f] = v;
        }
        validb[gp] = isrep ? 1 : 0;
    }
}

// ------------------------------------------------------------------
// Fused EdgeConv: block = 16 edge-rows (= 4 nodes x 4 edges), 512 thr
// (16 waves). Phase 1: build A=[x_i, x_j-x_i] tile in LDS, WMMA GEMM
// vs w1 (KA=2F), bias+lrelu -> H tile in LDS. Phase 2: WMMA GEMM vs
// w2 (K=C1), bias+lrelu, in-register reduce over the 4 edges of each
// node, store into the concat buffer slice (stride 772).
// fp32 WMMA: V_WMMA_F32_16X16X4_F32, exact fp32 semantics.
//
// Out-of-range columns: B column n only contributes to C column n, so
// instead of zeroing (which caused per-iteration exec divergence) we
// clamp the address; garbage columns are simply never stored.
// ------------------------------------------------------------------
__global__ __launch_bounds__(512)
void edgeconv_kernel(const float* xin, int xStride, int xOff, int F,
                     const int* __restrict__ nbr,
                     const float* __restrict__ w1, const float* __restrict__ b1,
                     const float* __restrict__ w2, const float* __restrict__ b2,
                     int C1, float* out, int oStride, int oOff)
{
    extern __shared__ float smem[];
    const int KA = 2*F, sA = KA + 1, sH = C1 + 1;
    float* As = smem;            // 16 x sA
    float* Hs = smem + 16*sA;    // 16 x sH
    const int tid = threadIdx.x;
    const int lane = tid & 31, wid = tid >> 5;
    const int rowBase = blockIdx.x * 16;

    // Build A tile: row r = edge rowBase+r, cols [x_i | x_j - x_i]
    for (int idx = tid; idx < 16*KA; idx += blockDim.x) {
        int r = idx / KA, c = idx % KA;
        int e = rowBase + r;
        int i = e >> 2;
        float v;
        if (c < F) {
            v = xin[(size_t)i*xStride + xOff + c];
        } else {
            int j = (i & ~(P_-1)) + nbr[e];
            int f = c - F;
            v = xin[(size_t)j*xStride + xOff + f] - xin[(size_t)i*xStride + xOff + f];
        }
        As[r*sA + c] = v;
    }
    __syncthreads();

    const int m   = lane & 15;            // A row / B col within tile
    const int khi = (lane >> 4) << 1;     // K offset: 0 or 2
    const int mhi = (lane >> 4) << 3;     // C row offset: 0 or 8

    // Phase 1: Hs = lrelu(A @ w1 + b1)
    {
        int col = wid*16 + m;
        bool cok = col < C1;
        int colc = cok ? col : (C1 - 1);   // clamp: no divergence in loop
        v8f acc = {};
        #pragma unroll 4
        for (int kk = 0; kk < KA; kk += 4) {
            v2f a, b;
            a.x = As[m*sA + kk + khi];
            a.y = As[m*sA + kk + khi + 1];
            b.x = w1[(kk + khi) * C1 + colc];
            b.y = w1[(kk + khi + 1) * C1 + colc];
            acc = __builtin_amdgcn_wmma_f32_16x16x4_f32(false, a, false, b, (short)0, acc, false, false);
        }
        if (cok) {
            float bias = b1[col];
            #pragma unroll
            for (int v = 0; v < 8; ++v)
                Hs[(v + mhi)*sH + col] = lrelu(acc[v] + bias);
        }
    }
    __syncthreads();

    // Phase 2: out_node = sum_k lrelu(H @ w2 + b2)
    {
        int col = wid*16 + m;
        bool cok = col < C2_;
        int colc = cok ? col : (C2_ - 1);
        v8f acc = {};
        #pragma unroll 4
        for (int kk = 0; kk < C1; kk += 4) {
            v2f a, b;
            a.x = Hs[m*sH + kk + khi];
            a.y = Hs[m*sH + kk + khi + 1];
            b.x = w2[(kk + khi) * C2_ + colc];
            b.y = w2[(kk + khi + 1) * C2_ + colc];
            acc = __builtin_amdgcn_wmma_f32_16x16x4_f32(false, a, false, b, (short)0, acc, false, false);
        }
        if (cok) {
            float bias = b2[col];
            float s0 = lrelu(acc[0]+bias) + lrelu(acc[1]+bias) + lrelu(acc[2]+bias) + lrelu(acc[3]+bias);
            float s1 = lrelu(acc[4]+bias) + lrelu(acc[5]+bias) + lrelu(acc[6]+bias) + lrelu(acc[7]+bias);
            int nodeBase = (rowBase >> 2) + ((lane < 16) ? 0 : 2);
            out[(size_t)nodeBase      *oStride + oOff + col] = s0;
            out[(size_t)(nodeBase + 1)*oStride + oOff + col] = s1;
        }
    }
}

// ------------------------------------------------------------------
// Generic node GEMM (n1, n2): 16 rows x 128 cols per block (8 waves).
// ------------------------------------------------------------------
__global__ __launch_bounds__(256)
void gemm_node_kernel(const float* in, int iStride, int iOff, int Kd,
                      const float* __restrict__ W, const float* __restrict__ bias,
                      int Ncol, int doLrelu,
                      float* out, int oStride, int oOff)
{
    extern __shared__ float smem[];
    const int sA = Kd + 1;
    float* As = smem;
    const int tid = threadIdx.x, lane = tid & 31, wid = tid >> 5;
    const int rowBase = blockIdx.x * 16;

    for (int idx = tid; idx < 16*Kd; idx += blockDim.x) {
        int r = idx / Kd, c = idx % Kd;
        As[r*sA + c] = in[(size_t)(rowBase + r)*iStride + iOff + c];
    }
    __syncthreads();

    const int m   = lane & 15;
    const int khi = (lane >> 4) << 1;
    const int mhi = (lane >> 4) << 3;
    int col = blockIdx.y*128 + wid*16 + m;
    bool cok = col < Ncol;
    int colc = cok ? col : (Ncol - 1);    // clamp: no divergence in loop
    v8f acc = {};
    #pragma unroll 4
    for (int kk = 0; kk < Kd; kk += 4) {
        v2f a, b;
        a.x = As[m*sA + kk + khi];
        a.y = As[m*sA + kk + khi + 1];
        b.x = W[(size_t)(kk + khi)*Ncol + colc];
        b.y = W[(size_t)(kk + khi + 1)*Ncol + colc];
        acc = __builtin_amdgcn_wmma_f32_16x16x4_f32(false, a, false, b, (short)0, acc, false, false);
    }
    if (cok) {
        float bv = bias[col];
        #pragma unroll
        for (int v = 0; v < 8; ++v) {
            float val = acc[v] + bv;
            if (doLrelu) val = lrelu(val);
            out[(size_t)(rowBase + v + mhi)*oStride + oOff + col] = val;
        }
    }
}

// ------------------------------------------------------------------
// Masked global pooling: max/min/sum/mean over valid nodes, lrelu.
// ------------------------------------------------------------------
__global__ void pool_kernel(const float* __restrict__ h, const int* __restrict__ valid,
                            float* __restrict__ pooled)
{
    int g = blockIdx.x, c = threadIdx.x;   // 192 threads
    float mx = -__builtin_inff(), mn = __builtin_inff(), sm = 0.f;
    int cnt = 0;
    for (int p = 0; p < P_; ++p) {
        int gp = g*P_ + p;
        if (valid[gp]) {
            float v = h[(size_t)gp*C2_ + c];
            mx = fmaxf(mx, v); mn = fminf(mn, v); sm += v; ++cnt;
        }
    }
    float mean = sm / (float)cnt;
    pooled[g*768 + c]       = lrelu(mx);
    pooled[g*768 + 192 + c] = lrelu(mn);
    pooled[g*768 + 384 + c] = lrelu(sm);
    pooled[g*768 + 576 + c] = lrelu(mean);
}

// ------------------------------------------------------------------
// Head MLP: n3 (768->96, lrelu) then n4 (96->1).
// ------------------------------------------------------------------
__global__ void head_kernel(const float* __restrict__ pooled,
                            const float* __restrict__ n3w, const float* __restrict__ n3b,
                            const float* __restrict__ n4w, const float* __restrict__ n4b,
                            float* __restrict__ out)
{
    __shared__ float h3[96];
    int g = blockIdx.x, t = threadIdx.x;   // 96 threads
    float acc = n3b[t];
    for (int i = 0; i < 768; ++i) acc += pooled[g*768 + i] * n3w[i*96 + t];
    h3[t] = lrelu(acc);
    __syncthreads();
    if (t == 0) {
        float a2 = n4b[0];
        for (int i = 0; i < 96; ++i) a2 += h3[i] * n4w[i];
        out[g] = a2;
    }
}

extern "C" void kernel_launch(void* const* d_in, const int* in_sizes, int n_in,
                              void* d_out, int out_size, void* d_ws, size_t ws_size,
                              hipStream_t stream)
{
    (void)in_sizes; (void)n_in; (void)out_size; (void)ws_size;
    const float* x       = (const float*)d_in[0];
    /* d_in[1] = batch (implicit in layout, unused) */
    const float* randpos = (const float*)d_in[2];
    const float* ep_w = (const float*)d_in[3];
    const float* ep_b = (const float*)d_in[4];
    const float* cw1[4] = { (const float*)d_in[5],  (const float*)d_in[9],
                            (const float*)d_in[13], (const float*)d_in[17] };
    const float* cb1[4] = { (const float*)d_in[6],  (const float*)d_in[10],
                            (const float*)d_in[14], (const float*)d_in[18] };
    const float* cw2[4] = { (const float*)d_in[7],  (const float*)d_in[11],
                            (const float*)d_in[15], (const float*)d_in[19] };
    const float* cb2[4] = { (const float*)d_in[8],  (const float*)d_in[12],
                            (const float*)d_in[16], (const float*)d_in[20] };
    const float* n1w = (const float*)d_in[21]; const float* n1b = (const float*)d_in[22];
    const float* n2w = (const float*)d_in[23]; const float* n2b = (const float*)d_in[24];
    const float* n3w = (const float*)d_in[25]; const float* n3b = (const float*)d_in[26];
    const float* n4w = (const float*)d_in[27]; const float* n4b = (const float*)d_in[28];
    float* outp = (float*)d_out;

    char* wsp = (char*)d_ws;
    size_t off = 0;
    auto alloc = [&](size_t bytes) -> char* {
        char* p = wsp + off; off += (bytes + 255) & ~(size_t)255; return p;
    };
    int*   nbr0   = (int*)  alloc((size_t)NK_ * sizeof(int));
    int*   nbr    = (int*)  alloc((size_t)NK_ * sizeof(int));
    float* escore = (float*)alloc((size_t)NK_ * sizeof(float));
    int*   validb = (int*)  alloc((size_t)N_  * sizeof(int));
    float* hcat   = (float*)alloc((size_t)N_ * HCAT_ * sizeof(float)); // [x2|a|b|c|d]
    float* hnode  = (float*)alloc((size_t)N_ * 252   * sizeof(float));
    float* hfin   = (float*)alloc((size_t)N_ * C2_   * sizeof(float));
    float* pooled = (float*)alloc((size_t)G_ * 768   * sizeof(float));

    // 1) KNN on random 2D positions
    knn_kernel<2><<<G_, P_, 0, stream>>>(randpos, 2, 0, nullptr, nbr0);
    // 2) EdgePooling edge scores (softmax over K per target, +0.5)
    ep_score_kernel<<<N_/256, 256, 0, stream>>>(x, nbr0, ep_w, ep_b, escore);
    // 3) Greedy merge -> hcat cols 0..3 + valid mask
    ep_match_kernel<<<G_, 1024, 0, stream>>>(x, nbr0, escore, hcat, validb);

    // 4) Four EdgeConv layers, fused GEMM1+GEMM2+K-reduce via fp32 WMMA
    struct LayerCfg { int F, C1, inOff, outOff; };
    const LayerCfg L[4] = {
        {   4,  96,   0,   4 },
        { 192, 252,   4, 196 },
        { 192, 252, 196, 388 },
        { 192, 252, 388, 580 },
    };
    for (int l = 0; l < 4; ++l) {
        knn_kernel<3><<<G_, P_, 0, stream>>>(hcat, HCAT_, L[l].inOff, validb, nbr);
        size_t smem = (size_t)(16*(2*L[l].F + 1) + 16*(L[l].C1 + 1)) * sizeof(float);
        edgeconv_kernel<<<NK_/16, 512, smem, stream>>>(
            hcat, HCAT_, L[l].inOff, L[l].F, nbr,
            cw1[l], cb1[l], cw2[l], cb2[l], L[l].C1,
            hcat, HCAT_, L[l].outOff);
    }

    // 5) n1: [N,772] @ [772,252] + lrelu
    {
        size_t smem = (size_t)16 * (HCAT_ + 1) * sizeof(float);
        dim3 grid(N_/16, 2);
        gemm_node_kernel<<<grid, 256, smem, stream>>>(hcat, HCAT_, 0, HCAT_,
                                                      n1w, n1b, 252, 1, hnode, 252, 0);
    }
    // 6) n2: [N,252] @ [252,192] (no activation)
    {
        size_t smem = (size_t)16 * (252 + 1) * sizeof(float);
        dim3 grid(N_/16, 2);
        gemm_node_kernel<<<grid, 256, smem, stream>>>(hnode, 252, 0, 252,
                                                      n2w, n2b, 192, 0, hfin, 192, 0);
    }
    // 7) masked pool4cat + lrelu
    pool_kernel<<<G_, 192, 0, stream>>>(hfin, validb, pooled);
    // 8) head MLP -> d_out [G]
    head_kernel<<<G_, 96, 0, stream>>>(pooled, n3w, n3b, n4w, n4b, outp);
}